// MotifPooling_55619826483394
// MI455X (gfx1250) — compile-verified
//
#include <hip/hip_runtime.h>
#include <math.h>

// Problem constants (fixed by the reference: H=256, GROUP=32, out = [G, 3H] f32)
#define GROUP     32
#define HDIM      256
#define ROWSTRIDE 264   // floats; 264 % 64 == 8 -> bank-conflict-free row rotation
#define OUTW      (3 * HDIM)

typedef __attribute__((ext_vector_type(2))) float v2f;
typedef __attribute__((ext_vector_type(4))) float v4f;
typedef __attribute__((ext_vector_type(8))) float v8f;

__global__ __launch_bounds__(256) void motif_pool_kernel(
    const float* __restrict__ h,      // [N, 256]
    const float* __restrict__ att_w,  // [256, 1]
    const float* __restrict__ att_b,  // [1]
    const int*   __restrict__ idx,    // [T]
    float*       __restrict__ out)    // [G, 768] = [mean | max | att]
{
    __shared__ float rows[GROUP * ROWSTRIDE];   // staged gathered rows
    __shared__ float wlds[HDIM];                // att_w
    __shared__ float logits[GROUP];
    __shared__ float wsm[GROUP];                // softmax weights
    __shared__ int   ridx[GROUP];

    const int g    = blockIdx.x;
    const int tid  = threadIdx.x;
    const int lane = tid & 31;
    const int wv   = tid >> 5;          // wave id 0..7 (wave32)

    // ---- phase 0: indices + attention vector into LDS -------------------
    if (tid < GROUP) ridx[tid] = idx[g * GROUP + tid];
    wlds[tid] = att_w[tid];
    const float bias = att_b[0];
    __syncthreads();

    // ---- phase 1: async gather of 32 rows (32 KB) into LDS --------------
    // wave wv copies rows 4wv..4wv+3; each row = 1024B = 2 x (32 lanes x 16B)
    {
        const int r0 = wv * 4;
        #pragma unroll
        for (int rr = 0; rr < 4; ++rr) {
            const int r = r0 + rr;
            const float* gsrc = h + (size_t)ridx[r] * HDIM;
            #pragma unroll
            for (int half = 0; half < 2; ++half) {
                unsigned long long ga =
                    (unsigned long long)(gsrc + half * 128 + lane * 4);
                unsigned lo =
                    (unsigned)(size_t)&rows[r * ROWSTRIDE + half * 128 + lane * 4];
                asm volatile("global_load_async_to_lds_b128 %0, %1, off"
                             :: "v"(lo), "v"(ga) : "memory");
            }
        }
        asm volatile("s_wait_asynccnt 0" ::: "memory");
    }
    __syncthreads();

    // ---- phase 2: logits[r] = rows[r] . att_w + b -----------------------
    // wave wv computes rows 4wv..4wv+3; 32 lanes strided partial dot.
    {
        #pragma unroll
        for (int rr = 0; rr < 4; ++rr) {
            const int r = wv * 4 + rr;
            float p = 0.0f;
            #pragma unroll
            for (int i = 0; i < 8; ++i) {
                const int k = lane + 32 * i;
                p += rows[r * ROWSTRIDE + k] * wlds[k];
            }
            #pragma unroll
            for (int off = 16; off; off >>= 1) p += __shfl_xor(p, off, 32);
            if (lane == 0) logits[r] = p + bias;
        }
    }
    __syncthreads();

    // ---- phase 3: stable 32-wide softmax (wave 0) -----------------------
    if (wv == 0) {
        float l = logits[lane];
        float mv = l;
        #pragma unroll
        for (int off = 16; off; off >>= 1) mv = fmaxf(mv, __shfl_xor(mv, off, 32));
        float e = expf(l - mv);
        float s = e;
        #pragma unroll
        for (int off = 16; off; off >>= 1) s += __shfl_xor(s, off, 32);
        wsm[lane] = e / s;
    }
    __syncthreads();

    // ---- phase 4: per-feature max (VALU column scan) --------------------
    {
        float mx = -INFINITY;
        #pragma unroll
        for (int r = 0; r < GROUP; ++r)
            mx = fmaxf(mx, rows[r * ROWSTRIDE + tid]);
        out[(size_t)g * OUTW + HDIM + tid] = mx;
    }

    // ---- phase 5: mean + att via v_wmma_f32_16x16x4_f32 -----------------
    // D = A(16x32) x B(32x256-tile), accumulated over K=32 in 8 chunks of 4.
    // A row 0 = 1/32 (mean), row 1 = softmax weights, rows 2..15 = 0.
    // Operand packing (ISA 7.12.2, f32 16x16x4):
    //   A: lanes 0-15 hold A[m, 0..1] in (x,y); lanes 16-31 hold A[m, 2..3]
    //   B: lanes 0-15 hold B[0..1, n] in (x,y); lanes 16-31 hold B[2..3, n]
    //   D: row M in VGPR M, lanes 0-15 (N=0..15)
    {
        const int  m  = lane & 15;           // A-row / B-col within tile
        const bool hi = (lane >= 16);
        const float MW = 1.0f / (float)GROUP;

        #pragma unroll
        for (int t = 0; t < 2; ++t) {        // wave wv owns feature tiles 2wv, 2wv+1
            const int tb = (wv * 2 + t) * 16;
            v8f c = {};
            #pragma unroll
            for (int rc = 0; rc < 8; ++rc) {
                const int rb = rc * 4 + (hi ? 2 : 0);
                const float s0 = wsm[rb];
                const float s1 = wsm[rb + 1];
                v2f a, b;
                a.x = (m == 0) ? MW : ((m == 1) ? s0 : 0.0f);
                a.y = (m == 0) ? MW : ((m == 1) ? s1 : 0.0f);
                b.x = rows[(rb    ) * ROWSTRIDE + tb + m];
                b.y = rows[(rb + 1) * ROWSTRIDE + tb + m];
                c = __builtin_amdgcn_wmma_f32_16x16x4_f32(
                        /*neg_a=*/false, a, /*neg_b=*/false, b,
                        /*c_mod=*/(short)0, c,
                        /*reuse_a=*/false, /*reuse_b=*/false);
            }
            if (lane < 16) {
                const size_t base = (size_t)g * OUTW;
                out[base +            tb + m] = c[0];   // M=0: mean
                out[base + 2 * HDIM + tb + m] = c[1];   // M=1: att
            }
        }
    }
}

extern "C" void kernel_launch(void* const* d_in, const int* in_sizes, int n_in,
                              void* d_out, int out_size, void* d_ws, size_t ws_size,
                              hipStream_t stream) {
    (void)in_sizes; (void)n_in; (void)d_ws; (void)ws_size;
    const float* h     = (const float*)d_in[0];
    const float* att_w = (const float*)d_in[1];
    const float* att_b = (const float*)d_in[2];
    const int*   idx   = (const int*)d_in[3];
    // d_in[4] (seg_ids) unused: segments are contiguous blocks of 32.
    // d_in[5] (num_groups) recoverable from out_size.
    const int G = out_size / OUTW;
    float* out = (float*)d_out;
    motif_pool_kernel<<<dim3(G), dim3(256), 0, stream>>>(h, att_w, att_b, idx, out);
}